// graph_gru_gcn_26508538151352
// MI455X (gfx1250) — compile-verified
//
#include <hip/hip_runtime.h>

#define N_NODES 50000
#define N_EDGES 800000
#define D 96
#define NLAYER 2
#define ND (N_NODES * D)

typedef __attribute__((ext_vector_type(2))) float v2f;
typedef __attribute__((ext_vector_type(8))) float v8f;

// ---------------------------------------------------------------------------
// GEMM: Y[N,96] = X[N,96] @ W[96,96]  (row-major), fp32 WMMA 16x16x4.
// Block = 192 threads = 6 waves; wave w computes the 16x16 tile at
// (rows 16*blockIdx.x, cols 16*w). N = 50000 = 3125*16, D = 96 = 6*16,
// so no bounds guards -> EXEC all ones, as WMMA requires.
//
// Lane layouts per CDNA5 ISA 7.12.2:
//   A 16x4 f32 : lane&15 = M,  half = lane>>4, vgpr v -> K = 2*half + v
//   B  4x16 f32: lane&15 = N,  half = lane>>4, vgpr v -> K = 2*half + v
//   C 16x16 f32: lane&15 = N,  vgpr r -> M = r + 8*half
// ---------------------------------------------------------------------------
__global__ void __launch_bounds__(192)
gemm96_wmma(const float* __restrict__ X, const float* __restrict__ W,
            float* __restrict__ Y) {
  const int lane = threadIdx.x & 31;
  const int wave = threadIdx.x >> 5;   // 0..5 -> column tile
  const int m    = lane & 15;
  const int half = lane >> 4;
  const int row0 = blockIdx.x << 4;
  const int col0 = wave << 4;

  const float* __restrict__ xrow = X + (row0 + m) * D;  // A-row for this lane
  const float* __restrict__ wcol = W + col0 + m;        // B-col for this lane

  v8f acc = {};
#pragma unroll
  for (int kb = 0; kb < D; kb += 4) {
    const int k = kb + 2 * half;
    v2f a, b;
    a.x = xrow[k];
    a.y = xrow[k + 1];
    b.x = wcol[k * D];
    b.y = wcol[(k + 1) * D];
    acc = __builtin_amdgcn_wmma_f32_16x16x4_f32(false, a, false, b,
                                                (short)0, acc, false, false);
  }

  float* __restrict__ yout = Y + col0 + m;   // n = lane&15
#pragma unroll
  for (int r = 0; r < 8; ++r) {
    yout[(row0 + r + 8 * half) * D] = acc[r];
  }
}

// ---------------------------------------------------------------------------
// Degree / normalization
// ---------------------------------------------------------------------------
__global__ void deg_init(float* __restrict__ deg) {
  int i = blockIdx.x * blockDim.x + threadIdx.x;
  if (i < N_NODES) deg[i] = 2.0f;             // improved self-loop weight 2
}

__global__ void deg_accum(const int* __restrict__ dst, float* __restrict__ deg) {
  int e = blockIdx.x * blockDim.x + threadIdx.x;
  if (e < N_EDGES) atomicAdd(&deg[dst[e]], 1.0f);
}

__global__ void deg_rsqrt(float* __restrict__ dis) {
  int i = blockIdx.x * blockDim.x + threadIdx.x;
  if (i < N_NODES) dis[i] = rsqrtf(dis[i]);
}

// ---------------------------------------------------------------------------
// agg[i,:] = 2*dis[i]^2 * xw[i,:]   (self-loop term, also initializes buffer)
// block (96,4)
// ---------------------------------------------------------------------------
__global__ void selfloop_init(float* __restrict__ agg, const float* __restrict__ xw,
                              const float* __restrict__ dis) {
  int i = blockIdx.x * blockDim.y + threadIdx.y;
  if (i >= N_NODES) return;
  float c = 2.0f * dis[i] * dis[i];
  int d = threadIdx.x;
  agg[i * D + d] = c * xw[i * D + d];
}

// ---------------------------------------------------------------------------
// agg[dst,:] += dis[src]*dis[dst] * xw[src,:] over all edges.
// block (24,16): 24 threads/edge, float4 per thread, atomics land in L2.
// ---------------------------------------------------------------------------
__global__ void edge_agg(const float* __restrict__ xw, const float* __restrict__ dis,
                         const int* __restrict__ src, const int* __restrict__ dst,
                         float* __restrict__ agg) {
  int e = blockIdx.x * blockDim.y + threadIdx.y;
  if (e >= N_EDGES) return;
  int s = src[e], t = dst[e];
  float nrm = dis[s] * dis[t];
  int d = threadIdx.x * 4;
  const float4 v = *(const float4*)(xw + s * D + d);
  float* o = agg + t * D + d;
  atomicAdd(o + 0, v.x * nrm);
  atomicAdd(o + 1, v.y * nrm);
  atomicAdd(o + 2, v.z * nrm);
  atomicAdd(o + 3, v.w * nrm);
}

// ---------------------------------------------------------------------------
// Elementwise GRU pieces
// ---------------------------------------------------------------------------
__global__ void ew_sigmoid_sum(float* __restrict__ a, const float* __restrict__ b) {
  int i = blockIdx.x * blockDim.x + threadIdx.x;
  if (i >= ND) return;
  float v = fmaxf(a[i], 0.0f) + fmaxf(b[i], 0.0f);
  a[i] = 1.0f / (1.0f + __expf(-v));
}

__global__ void ew_tanh_sum(float* __restrict__ a, const float* __restrict__ b) {
  int i = blockIdx.x * blockDim.x + threadIdx.x;
  if (i >= ND) return;
  float v = fmaxf(a[i], 0.0f) + fmaxf(b[i], 0.0f);
  a[i] = tanhf(v);
}

__global__ void ew_mul(float* __restrict__ o, const float* __restrict__ a,
                       const float* __restrict__ b) {
  int i = blockIdx.x * blockDim.x + threadIdx.x;
  if (i >= ND) return;
  o[i] = a[i] * b[i];
}

__global__ void ew_gru_out(float* __restrict__ o, const float* __restrict__ z,
                           const float* __restrict__ h, const float* __restrict__ ht) {
  int i = blockIdx.x * blockDim.x + threadIdx.x;
  if (i >= ND) return;
  float zi = z[i];
  o[i] = zi * h[i] + (1.0f - zi) * ht[i];
}

// ---------------------------------------------------------------------------
// Host orchestration
// ---------------------------------------------------------------------------
extern "C" void kernel_launch(void* const* d_in, const int* in_sizes, int n_in,
                              void* d_out, int out_size, void* d_ws, size_t ws_size,
                              hipStream_t stream) {
  const float* inp  = (const float*)d_in[0];
  const int*   ei   = (const int*)d_in[1];          // [2, E] int32 (JAX x64 off)
  const float* hbuf = (const float*)d_in[2];        // [2, N, D]
  const float* W_xz = (const float*)d_in[3];
  const float* W_hz = (const float*)d_in[4];
  const float* W_xr = (const float*)d_in[5];
  const float* W_hr = (const float*)d_in[6];
  const float* W_xh = (const float*)d_in[7];
  const float* W_hh = (const float*)d_in[8];
  float* out = (float*)d_out;                       // [2*2, N, D] (tuple dup)

  const int* src = ei;
  const int* dst = ei + N_EDGES;

  float* ws  = (float*)d_ws;
  float* dis = ws;                 // N
  float* xw  = dis + N_NODES;      // N*D  (GEMM scratch)
  float* t1  = xw + ND;            // N*D
  float* t2  = t1 + ND;
  float* t3  = t2 + ND;
  float* t4  = t3 + ND;

  const dim3 bSelf(96, 4), gSelf((N_NODES + 3) / 4);
  const dim3 bEdge(24, 16), gEdge((N_EDGES + 15) / 16);
  const int gEw = (ND + 255) / 256;

  // --- degree / normalization (dis buffer doubles as deg accumulator) ---
  deg_init<<<(N_NODES + 255) / 256, 256, 0, stream>>>(dis);
  deg_accum<<<(N_EDGES + 255) / 256, 256, 0, stream>>>(dst, dis);
  deg_rsqrt<<<(N_NODES + 255) / 256, 256, 0, stream>>>(dis);

  // one GCN conv: dest = (sym-norm A + 2I) @ (x @ W)    (relu applied later)
  auto conv = [&](const float* x, const float* Wm, float* dest) {
    gemm96_wmma<<<N_NODES / 16, 192, 0, stream>>>(x, Wm, xw);
    selfloop_init<<<gSelf, bSelf, 0, stream>>>(dest, xw, dis);
    edge_agg<<<gEdge, bEdge, 0, stream>>>(xw, dis, src, dst, dest);
  };

  for (int L = 0; L < NLAYER; ++L) {
    const float* x  = (L == 0) ? inp : (out + (L - 1) * ND);
    const float* hl = hbuf + L * ND;
    const size_t wo = (size_t)L * D * D;

    // z gate
    conv(x,  W_xz + wo, t1);
    conv(hl, W_hz + wo, t2);
    ew_sigmoid_sum<<<gEw, 256, 0, stream>>>(t1, t2);          // t1 = z

    // r gate
    conv(x,  W_xr + wo, t2);
    conv(hl, W_hr + wo, t3);
    ew_sigmoid_sum<<<gEw, 256, 0, stream>>>(t2, t3);          // t2 = r

    // candidate
    conv(x, W_xh + wo, t3);                                   // t3 = gcn(x,Wxh)
    ew_mul<<<gEw, 256, 0, stream>>>(t4, t2, hl);              // t4 = r*h
    conv(t4, W_hh + wo, t4);                                  // gemm reads t4 first
    ew_tanh_sum<<<gEw, 256, 0, stream>>>(t3, t4);             // t3 = h_tilde

    // h_out = z*h + (1-z)*h_tilde
    ew_gru_out<<<gEw, 256, 0, stream>>>(out + L * ND, t1, hl, t3);
  }

  // reference returns (h_out, h_out): duplicate first half into second half
  hipMemcpyAsync(out + (size_t)NLAYER * ND, out,
                 (size_t)NLAYER * ND * sizeof(float),
                 hipMemcpyDeviceToDevice, stream);
}